// DynamicRoutingOld_79353815761243
// MI455X (gfx1250) — compile-verified
//
#include <hip/hip_runtime.h>
#include <cstdint>
#include <cmath>

// ---------------------------------------------------------------------------
// Dynamic routing (capsule nets), B=128, J=32, I=1152, N=16, 3 iterations.
//
// Roofline: u_hat = 302MB f32, total FLOPs < 1 GFLOP -> pure HBM-bandwidth
// problem on MI455X (23.3 TB/s). The pipeline streams u_hat exactly 3x (one
// pass per routing iteration, ~0.9GB ~= 39us at roofline) by fusing the
// agreement update, per-(b,i)-column top-k mask/softmax/entropy, and the
// partial s accumulation into one kernel per iteration.
//
// Matrix ops use V_WMMA_F32_16X16X4_F32 (full f32, K=4 contraction).
// Operand streams are software-pipelined (4-deep) so loads stay in flight
// while WMMAs retire; v staging uses global_load_async_to_lds_b128 +
// s_wait_asynccnt; streaming loads are non-temporal; tiles get
// global_prefetch_b8.
// ---------------------------------------------------------------------------

#define JCAPS 32
#define ICAPS 1152
#define NCAPS 16
#define BSZ   128

typedef float v2f __attribute__((ext_vector_type(2)));
typedef float v8f __attribute__((ext_vector_type(8)));

// Types matching async builtin "vV4i*1V4i*3IiIi":
// (int4 addrspace(1)* src, int4 addrspace(3)* dst, imm offset, imm cpol)
typedef int v4i_vs __attribute__((vector_size(16)));
typedef v4i_vs __attribute__((address_space(1))) g_v4i;   // global int4
typedef v4i_vs __attribute__((address_space(3))) l_v4i;   // LDS int4

// D = A(16x4) * B(4x16) + C, f32.
static __device__ __forceinline__ v8f wmma4(v2f a, v2f b, v8f c) {
  return __builtin_amdgcn_wmma_f32_16x16x4_f32(false, a, false, b, (short)0, c,
                                               false, false);
}

// ---------------------------------------------------------------------------
// Iteration 0: b=0 -> c = 1/32 uniform, ent = ln(32).
// s0[b,j,n] = (1/32) * sum_i u[b,j,i,n]; then bias+reset+squash -> v0.
// One (b,j) per block; 4 waves split the i range; WMMA with A == 1/32
// broadcast does the K=4 column-sum chunks. B-operand stream is pipelined
// 4 chunks deep so HBM latency overlaps the WMMA chain.
// ---------------------------------------------------------------------------
__global__ void s0_kernel(const float* __restrict__ u,
                          const float* __restrict__ bias,
                          float* __restrict__ v_out) {
  const int e    = blockIdx.x;            // b*J + j
  const int j    = e & (JCAPS - 1);
  const int lane = threadIdx.x & 31;
  const int wave = threadIdx.x >> 5;      // 0..3
  const int half = lane >> 4;             // ISA layout: lanes 16-31 hold K2/K3
  const int col  = lane & 15;

  const float* ub = u + (size_t)e * ICAPS * NCAPS;

  v8f acc0 = {}; v8f acc1 = {};
  const v2f a = {0.03125f, 0.03125f};     // c = 1/32 everywhere
  const int iBeg = wave * (ICAPS / 4);    // 288 i-rows per wave
  const int iEnd = iBeg + (ICAPS / 4);

  // B(4x16): VGPR0 = row K0|K2, VGPR1 = row K1|K3; row k = u row (i+k)
  const int boff0 = (half ? 32 : 0) + col;
  const int boff1 = boff0 + 16;

  v2f buf[4];                             // 4-deep pipelined operand buffer
#pragma unroll
  for (int q = 0; q < 4; ++q) {
    const float* p = ub + (size_t)(iBeg + q * 4) * NCAPS;
    buf[q].x = __builtin_nontemporal_load(p + boff0);
    buf[q].y = __builtin_nontemporal_load(p + boff1);
  }
  for (int i = iBeg; i < iEnd; i += 16) { // 18 iterations of 4 WMMAs
    __builtin_prefetch(ub + (size_t)((i + 32 < iEnd) ? i + 32 : iBeg) * NCAPS,
                       0, 1);
#pragma unroll
    for (int q = 0; q < 4; ++q) {
      v2f cur = buf[q];
      int inext = i + 16 + q * 4;
      if (inext >= iEnd) inext = iBeg;    // dummy refill on final iteration
      const float* pn = ub + (size_t)inext * NCAPS;
      buf[q].x = __builtin_nontemporal_load(pn + boff0);
      buf[q].y = __builtin_nontemporal_load(pn + boff1);
      if (q & 1) acc1 = wmma4(a, cur, acc1);
      else       acc0 = wmma4(a, cur, acc0);
    }
  }

  __shared__ float sred[4 * 16];
  if (lane < 16) sred[wave * 16 + lane] = acc0[0] + acc1[0];  // row M=0
  __syncthreads();

  if (threadIdx.x < 16) {
    const int n = threadIdx.x;
    float s = sred[n] + sred[16 + n] + sred[32 + n] + sred[48 + n];
    float sum_s = s;
#pragma unroll
    for (int m = 1; m < 16; m <<= 1) sum_s += __shfl_xor(sum_s, m);
    float x = s + bias[j * NCAPS + n];
    if (sum_s == 0.0f) x = 0.0f;          // reset all-zero rows (ref semantics)
    float sq = x * x;
#pragma unroll
    for (int m = 1; m < 16; m <<= 1) sq += __shfl_xor(sq, m);
    v_out[(size_t)e * NCAPS + n] =
        (sq > 0.0f) ? (sq / (1.0f + sq)) * x * rsqrtf(sq) : 0.0f;
  }
}

// ---------------------------------------------------------------------------
// Fused routing iteration: one pass over u_hat.
//  phase 1: bu[i][j] = sum_n u[b,j,i,n]*v[b,j,n]      (WMMA, B = v broadcast)
//  phase 2: per (b,i) column: t = prev_b + bu; bottom-9 mask (stable rank via
//           lane shuffles); softmax; entropy -> atomic stat; c -> LDS
//  phase 3: s[b,j,n] += sum_i c[i][j]*u[b,j,i,n]      (WMMA, A = c broadcast)
// Block = 256 threads (8 waves), tile = 32 i-columns, grid = (36, 128).
// All K-chunk operands are hoisted into registers before the WMMA chain so
// the loads issue as a clause and the WMMAs run back-to-back.
// ---------------------------------------------------------------------------
template <bool FIRST>
__global__ void route_iter_kernel(const float* __restrict__ u,
                                  const float* __restrict__ v_in,
                                  float* __restrict__ b_ws,
                                  float* __restrict__ s_acc,
                                  float* __restrict__ stat_out) {
  const int bb   = blockIdx.y;
  const int i0   = blockIdx.x * 32;
  const int tid  = threadIdx.x;
  const int lane = tid & 31;
  const int wave = tid >> 5;              // 0..7
  const int half = lane >> 4;
  const int col  = lane & 15;

  __shared__ __align__(16) float v_lds[JCAPS * NCAPS];  // 512 f32
  __shared__ float bu_lds[32 * JCAPS];                  // [i_local][j]
  __shared__ __align__(8) float c_lds[JCAPS * 32];      // [j][i_local]

  const float* ub = u + (size_t)bb * JCAPS * ICAPS * NCAPS;
  const float* vb = v_in + (size_t)bb * JCAPS * NCAPS;

  // Prefetch the block's 64KB streaming tile (global_prefetch_b8): the tile
  // is consumed twice (phase 1 + phase 3), so pull it into near caches.
  {
    const int jpf = tid >> 3;
    const char* pj = (const char*)(ub + ((size_t)jpf * ICAPS + i0) * NCAPS);
    __builtin_prefetch(pj + (tid & 7) * 256, 0, 3);
    __builtin_prefetch(pj + (tid & 7) * 256 + 128, 0, 3);
  }

  // Stage v (2KB) into LDS — gfx1250 async-to-LDS path when available.
#if __has_builtin(__builtin_amdgcn_global_load_async_to_lds_b128)
  if (tid < 128) {
    __builtin_amdgcn_global_load_async_to_lds_b128(
        (g_v4i*)(vb + tid * 4), (l_v4i*)(&v_lds[tid * 4]), 0, 0);
  }
#if __has_builtin(__builtin_amdgcn_s_wait_asynccnt)
  __builtin_amdgcn_s_wait_asynccnt(0);
#else
  asm volatile("s_wait_asynccnt 0" ::: "memory");
#endif
#else
  for (int idx = tid; idx < JCAPS * NCAPS; idx += 256) v_lds[idx] = vb[idx];
#endif
  __syncthreads();

  // ---- phase 1: agreement via WMMA (A = u rows, B = v broadcast over N) ----
  for (int jj = wave; jj < JCAPS; jj += 8) {
    const float* uj = ub + ((size_t)jj * ICAPS + i0) * NCAPS;
#pragma unroll
    for (int st = 0; st < 2; ++st) {      // two 16-row i sub-tiles
      const float* urow = uj + (size_t)(st * 16 + col) * NCAPS;  // M = col
      v2f A1[4], B1[4];
#pragma unroll
      for (int c = 0; c < 4; ++c)         // adjacent pair -> one b64 load
        A1[c] = *(const v2f*)(urow + c * 4 + (half ? 2 : 0));
#pragma unroll
      for (int c = 0; c < 4; ++c)         // adjacent pair -> one ds_load_b64
        B1[c] = *(const v2f*)(&v_lds[jj * NCAPS + c * 4 + (half ? 2 : 0)]);
      v8f acc = {};
#pragma unroll
      for (int c = 0; c < 4; ++c) acc = wmma4(A1[c], B1[c], acc);
      // D[m, *] identical over columns; lane0 holds M=0..7, lane16 M=8..15
      if (col == 0) {
        const int ibase = st * 16 + (half ? 8 : 0);
#pragma unroll
        for (int r = 0; r < 8; ++r) bu_lds[(ibase + r) * JCAPS + jj] = acc[r];
      }
    }
  }
  __syncthreads();

  // ---- phase 2: per-column logits update / top-k mask / softmax / entropy --
  const float invBI = 1.0f / (float)(BSZ * ICAPS);
  for (int il = wave; il < 32; il += 8) {
    float t = bu_lds[il * JCAPS + lane];
    if (!FIRST) t += b_ws[((size_t)bb * ICAPS + (i0 + il)) * JCAPS + lane];
    // stable rank of t among the 32 j-values (ties broken by index)
    int r = 0;
    for (int m = 0; m < 32; ++m) {
      float y = __shfl(t, m);
      r += (y < t) || ((y == t) && (m < lane));
    }
    unsigned long long bal = __ballot(r == 8);   // mask_count-1 = 8
    float kth = __shfl(t, __ffsll(bal) - 1);     // 9th smallest (incl. -inf)
    t = (t <= kth) ? -__builtin_huge_valf() : t;
    if (FIRST)  // masked logits feed the next iteration's update
      b_ws[((size_t)bb * ICAPS + (i0 + il)) * JCAPS + lane] = t;
    float mx = t;
#pragma unroll
    for (int m = 16; m >= 1; m >>= 1) mx = fmaxf(mx, __shfl_xor(mx, m));
    float ev = __expf(t - mx);                   // exp(-inf)=0 for masked
    float sum = ev;
#pragma unroll
    for (int m = 16; m >= 1; m >>= 1) sum += __shfl_xor(sum, m);
    float c = ev / sum;
    c_lds[lane * 32 + il] = c;                   // [j][i] for phase-3 b64 reads
    float term = (c > 0.0f) ? -c * __logf(c) : 0.0f;
#pragma unroll
    for (int m = 16; m >= 1; m >>= 1) term += __shfl_xor(term, m);
    if (lane == 0) atomicAdd(stat_out, term * invBI);
  }
  __syncthreads();

  // ---- phase 3: partial s via WMMA (A = c broadcast over M, B = u rows) ----
  for (int jj = wave; jj < JCAPS; jj += 8) {
    const float* uj = ub + ((size_t)jj * ICAPS + i0) * NCAPS;
    v2f A3[8], B3[8];
#pragma unroll
    for (int s8 = 0; s8 < 8; ++s8) {      // hoist all operands (load clauses)
      const int k0 = s8 * 4 + (half ? 2 : 0);
      A3[s8] = *(const v2f*)(&c_lds[jj * 32 + k0]);  // ds_load_b64 broadcast
      B3[s8].x = uj[(size_t)k0 * NCAPS + col];
      B3[s8].y = uj[(size_t)(k0 + 1) * NCAPS + col];
    }
    v8f acc = {};
#pragma unroll
    for (int s8 = 0; s8 < 8; ++s8) acc = wmma4(A3[s8], B3[s8], acc);
    if (lane < 16)                        // row M=0: s_part[jj, n=lane]
      atomicAdd(&s_acc[((size_t)bb * JCAPS + jj) * NCAPS + lane], acc[0]);
  }
}

// ---------------------------------------------------------------------------
// bias + reset + squash:  v = (|s|^2/(1+|s|^2)) * s/|s|   (16 lanes per (b,j))
// ---------------------------------------------------------------------------
__global__ void squash_kernel(const float* __restrict__ s,
                              const float* __restrict__ bias,
                              float* __restrict__ out) {
  const int idx = blockIdx.x * 256 + threadIdx.x;   // 0..65535
  const int e   = idx >> 4;                         // b*J + j
  const int n   = idx & 15;
  const int j   = e & (JCAPS - 1);
  float sv = s[idx];
  float sum_s = sv;
#pragma unroll
  for (int m = 1; m < 16; m <<= 1) sum_s += __shfl_xor(sum_s, m);
  float x = sv + bias[j * NCAPS + n];
  if (sum_s == 0.0f) x = 0.0f;
  float sq = x * x;
#pragma unroll
  for (int m = 1; m < 16; m <<= 1) sq += __shfl_xor(sq, m);
  out[idx] = (sq > 0.0f) ? (sq / (1.0f + sq)) * x * rsqrtf(sq) : 0.0f;
}

__global__ void init_stats_kernel(float* stats) {
  if (threadIdx.x == 0) {
    stats[0] = logf(32.0f);   // iter0: uniform c=1/32 -> entropy exactly ln 32
    stats[1] = 0.0f;
    stats[2] = 0.0f;
  }
}

__global__ void zero_kernel(float* __restrict__ p, int n) {
  int i = blockIdx.x * 256 + threadIdx.x;
  if (i < n) p[i] = 0.0f;
}

// ---------------------------------------------------------------------------
extern "C" void kernel_launch(void* const* d_in, const int* in_sizes, int n_in,
                              void* d_out, int out_size, void* d_ws,
                              size_t ws_size, hipStream_t stream) {
  (void)in_sizes; (void)n_in; (void)out_size; (void)ws_size;

  const float* u    = (const float*)d_in[0];   // [128,32,1152,16] f32
  const float* bias = (const float*)d_in[1];   // [32,16] f32
  // d_in[2] = iters (=3): pipeline is specialized for 3 iterations.

  float* out_v = (float*)d_out;                 // [128,32,16]
  float* stats = out_v + (size_t)BSZ * JCAPS * NCAPS;  // 3 floats

  // Workspace (~18.6 MB): masked logits [b][i][j], v, s accumulator.
  float* b_ws = (float*)d_ws;                                  // B*I*J
  float* v_ws = b_ws + (size_t)BSZ * ICAPS * JCAPS;            // B*J*16
  float* s_ws = v_ws + (size_t)BSZ * JCAPS * NCAPS;            // B*J*16

  const int SJN = BSZ * JCAPS * NCAPS;          // 65536
  const dim3 fgrid(ICAPS / 32, BSZ);            // (36,128)

  init_stats_kernel<<<1, 32, 0, stream>>>(stats);
  zero_kernel<<<(SJN + 255) / 256, 256, 0, stream>>>(s_ws, SJN);

  // iter 0: u pass #1 -> v0
  s0_kernel<<<BSZ * JCAPS, 128, 0, stream>>>(u, bias, v_ws);
  // iter 1: u pass #2 (fused agreement+mask+softmax+partial-s) -> v1
  route_iter_kernel<true><<<fgrid, 256, 0, stream>>>(u, v_ws, b_ws, s_ws,
                                                     stats + 1);
  squash_kernel<<<SJN / 256, 256, 0, stream>>>(s_ws, bias, v_ws);
  zero_kernel<<<(SJN + 255) / 256, 256, 0, stream>>>(s_ws, SJN);
  // iter 2: u pass #3 -> v2 (final output)
  route_iter_kernel<false><<<fgrid, 256, 0, stream>>>(u, v_ws, b_ws, s_ws,
                                                      stats + 2);
  squash_kernel<<<SJN / 256, 256, 0, stream>>>(s_ws, bias, out_v);
}